// VMAESimCLR_12275016532038
// MI455X (gfx1250) — compile-verified
//
#include <hip/hip_runtime.h>
#include <hip/hip_bf16.h>
#include <math.h>

typedef float v2f __attribute__((ext_vector_type(2)));
typedef float v8f __attribute__((ext_vector_type(8)));

#define B_ROWS 4096
#define D_DIM  768
#define C_DIM  128
#define NCLS   8

#define WMMA_F32(a, b, c) \
    __builtin_amdgcn_wmma_f32_16x16x4_f32(false, (a), false, (b), (short)0, (c), false, false)

// ---------------------------------------------------------------------------
// Zero-init a float range
// ---------------------------------------------------------------------------
__global__ void zero_kernel(float* __restrict__ p, int n) {
    int i = blockIdx.x * blockDim.x + threadIdx.x;
    if (i < n) p[i] = 0.0f;
}

// ---------------------------------------------------------------------------
// LDS-staged WMMA f32 GEMM:  C[M,N] = A[M,K] * Bm[N,K]^T + bias[N]  (opt. ReLU)
//
// Block = 256 threads = 8 waves (4 along M x 2 along N), macro-tile 128x64.
// Each wave computes a 32x32 block as 2x2 WMMA tiles (V_WMMA_F32_16X16X4_F32).
// K is consumed in 32-wide panels, double-buffered in LDS:
//   - global loads for panel p+1 are issued into registers BEFORE the 8
//     k-steps of compute on panel p (a full panel of WMMAs hides the L2 trip),
//   - inner-loop operands come from LDS (ds_load_b64, short latency),
//   - ds_store of panel p+1 + one barrier per panel.
// LDS row stride = 36 floats (144B = 9*16B): float4 staging stays 16B-aligned,
// and fragment reads are bank-conflict-free (low lanes hit banks =0 mod 4,
// high lanes (k+2) hit banks =2 mod 4).
//
// Fragment layouts (CDNA5 ISA 7.12.2, 32-bit, wave32):
//   A 16x4 : lanes 0-15 -> M=lane, VGPR0/1 = K=0/1 ; lanes 16-31 -> K=2/3
//   B 4x16 : lanes 0-15 -> N=lane, VGPR0/1 = K=0/1 ; lanes 16-31 -> K=2/3
//   C 16x16: VGPR r, lanes 0-15 -> M=r, N=lane ; lanes 16-31 -> M=r+8
// ---------------------------------------------------------------------------
template <bool RELU>
__global__ __launch_bounds__(256, 1) void wmma_gemm_lds(
    const float* __restrict__ A,     // [M,K] row-major
    const float* __restrict__ Bm,    // [N,K] row-major
    const float* __restrict__ bias,  // [N]
    float* __restrict__ C,           // [M,N]
    int M, int N, int K) {
    constexpr int BM = 128, BN = 64, BK = 32, LDA = 36, LDB = 36;
    __shared__ float sA[2][BM * LDA];   // 2 x 18 KB
    __shared__ float sB[2][BN * LDB];   // 2 x  9 KB

    int tid = threadIdx.x;
    int tilesN = N / BN;
    int bm = blockIdx.x / tilesN, bn = blockIdx.x % tilesN;
    int rowBase = bm * BM, colBase = bn * BN;

    int wave = tid >> 5, lane = tid & 31;
    int wm = wave >> 1, wn = wave & 1;        // 4 (M) x 2 (N) wave grid
    int r  = lane & 15;
    int kh = (lane >> 4) << 1;                // 0 or 2

    // Staging assignment: A panel = 128x32 = 1024 float4s -> 4/thread,
    //                     B panel =  64x32 =  512 float4s -> 2/thread.
    int arow[4], akc[4], brow[2], bkc[2];
#pragma unroll
    for (int j = 0; j < 4; ++j) { int c = tid + 256 * j; arow[j] = c >> 3; akc[j] = (c & 7) << 2; }
#pragma unroll
    for (int j = 0; j < 2; ++j) { int c = tid + 256 * j; brow[j] = c >> 3; bkc[j] = (c & 7) << 2; }

    const float* gA = A  + (size_t)rowBase * K;
    const float* gB = Bm + (size_t)colBase * K;

    float4 ra[4], rb[2];
    // ---- prologue: stage panel 0 ----
#pragma unroll
    for (int j = 0; j < 4; ++j) ra[j] = *(const float4*)(gA + (size_t)arow[j] * K + akc[j]);
#pragma unroll
    for (int j = 0; j < 2; ++j) rb[j] = *(const float4*)(gB + (size_t)brow[j] * K + bkc[j]);
#pragma unroll
    for (int j = 0; j < 4; ++j) *(float4*)&sA[0][arow[j] * LDA + akc[j]] = ra[j];
#pragma unroll
    for (int j = 0; j < 2; ++j) *(float4*)&sB[0][brow[j] * LDB + bkc[j]] = rb[j];
    __syncthreads();

    v8f acc00 = {}, acc01 = {}, acc10 = {}, acc11 = {};

    int a0off = (wm * 32 +      r) * LDA + kh;
    int a1off = (wm * 32 + 16 + r) * LDA + kh;
    int b0off = (wn * 32 +      r) * LDB + kh;
    int b1off = (wn * 32 + 16 + r) * LDB + kh;

    const int P = K / BK;                     // 24 panels for K=768
#pragma unroll 2
    for (int p = 0; p < P; ++p) {
        const int cur = p & 1, nxt = cur ^ 1;
        if (p + 1 < P) {
            // issue global loads for panel p+1 (in flight across the compute)
            int kp = (p + 1) * BK;
#pragma unroll
            for (int j = 0; j < 4; ++j) ra[j] = *(const float4*)(gA + (size_t)arow[j] * K + kp + akc[j]);
#pragma unroll
            for (int j = 0; j < 2; ++j) rb[j] = *(const float4*)(gB + (size_t)brow[j] * K + kp + bkc[j]);
        }
        // compute 8 k-steps on panel p from LDS
        const float* pA = &sA[cur][0];
        const float* pB = &sB[cur][0];
#pragma unroll
        for (int kk = 0; kk < BK; kk += 4) {
            v2f a0 = *(const v2f*)(pA + a0off + kk);
            v2f a1 = *(const v2f*)(pA + a1off + kk);
            v2f b0 = *(const v2f*)(pB + b0off + kk);
            v2f b1 = *(const v2f*)(pB + b1off + kk);
            acc00 = WMMA_F32(a0, b0, acc00);
            acc01 = WMMA_F32(a0, b1, acc01);
            acc10 = WMMA_F32(a1, b0, acc10);
            acc11 = WMMA_F32(a1, b1, acc11);
        }
        if (p + 1 < P) {
#pragma unroll
            for (int j = 0; j < 4; ++j) *(float4*)&sA[nxt][arow[j] * LDA + akc[j]] = ra[j];
#pragma unroll
            for (int j = 0; j < 2; ++j) *(float4*)&sB[nxt][brow[j] * LDB + bkc[j]] = rb[j];
        }
        __syncthreads();
    }

    // ---- epilogue: bias (+ReLU) and store ----
    int row0  = rowBase + wm * 32;
    int col0  = colBase + wn * 32;
    int n0    = col0 + r;
    int rbase = row0 + ((lane >> 4) << 3);    // +0 (lanes 0-15) or +8 (16-31)
    float bv0 = bias[n0];
    float bv1 = bias[n0 + 16];
#pragma unroll
    for (int i = 0; i < 8; ++i) {
        float v00 = acc00[i] + bv0;
        float v01 = acc01[i] + bv1;
        float v10 = acc10[i] + bv0;
        float v11 = acc11[i] + bv1;
        if (RELU) {
            v00 = fmaxf(v00, 0.f); v01 = fmaxf(v01, 0.f);
            v10 = fmaxf(v10, 0.f); v11 = fmaxf(v11, 0.f);
        }
        C[(size_t)(rbase + i)      * N + n0]      = v00;
        C[(size_t)(rbase + i)      * N + n0 + 16] = v01;
        C[(size_t)(rbase + i + 16) * N + n0]      = v10;
        C[(size_t)(rbase + i + 16) * N + n0 + 16] = v11;
    }
}

// ---------------------------------------------------------------------------
// Per-row log-softmax of (2*embeds) + per-class accumulation.
// Block = 256 threads = 8 waves, each wave handles 8 rows (64 rows / block).
// ---------------------------------------------------------------------------
__global__ __launch_bounds__(256) void stats_kernel(
    const float* __restrict__ embeds, const int* __restrict__ labels,
    float* __restrict__ Pg, float* __restrict__ Lg,
    float* __restrict__ HSg, float* __restrict__ CNTg) {
    __shared__ float sP[NCLS * D_DIM];
    __shared__ float sL[NCLS * D_DIM];
    for (int i = threadIdx.x; i < NCLS * D_DIM; i += 256) { sP[i] = 0.f; sL[i] = 0.f; }
    __syncthreads();

    int wave = threadIdx.x >> 5;
    int lane = threadIdx.x & 31;
    int row0 = blockIdx.x * 64 + wave * 8;

    for (int rr = 0; rr < 8; ++rr) {
        int row = row0 + rr;
        const float* e = embeds + (size_t)row * D_DIM;
        float x[D_DIM / 32];
        float m = -INFINITY;
#pragma unroll
        for (int t = 0; t < D_DIM / 32; ++t) {
            x[t] = 2.0f * e[lane + 32 * t];   // embeds / TEMPERATURE, T = 0.5
            m = fmaxf(m, x[t]);
        }
        for (int off = 16; off >= 1; off >>= 1) m = fmaxf(m, __shfl_xor(m, off, 32));
        float s = 0.f;
#pragma unroll
        for (int t = 0; t < D_DIM / 32; ++t) s += expf(x[t] - m);
        for (int off = 16; off >= 1; off >>= 1) s += __shfl_xor(s, off, 32);
        float lse = m + logf(s);

        int c = labels[row] & (NCLS - 1);
        float hneg = 0.f;                     // sum p*logp = -H
#pragma unroll
        for (int t = 0; t < D_DIM / 32; ++t) {
            float lp = x[t] - lse;
            float p  = expf(lp);
            hneg += p * lp;
            int d = lane + 32 * t;
            atomicAdd(&sP[c * D_DIM + d], p);
            atomicAdd(&sL[c * D_DIM + d], lp);
        }
        for (int off = 16; off >= 1; off >>= 1) hneg += __shfl_xor(hneg, off, 32);
        if (lane == 0) {
            atomicAdd(&HSg[c], -hneg);        // H_i = -sum p*logp
            atomicAdd(&CNTg[c], 1.0f);
        }
    }
    __syncthreads();
    for (int i = threadIdx.x; i < NCLS * D_DIM; i += 256) {
        atomicAdd(&Pg[i], sP[i]);
        atomicAdd(&Lg[i], sL[i]);
    }
}

// ---------------------------------------------------------------------------
// Cross-entropy: one wave per 128-wide logits row.
// ---------------------------------------------------------------------------
__global__ __launch_bounds__(256) void ce_kernel(
    const float* __restrict__ logits, const int* __restrict__ labels,
    float* __restrict__ ceAcc) {
    int wave = (blockIdx.x * blockDim.x + threadIdx.x) >> 5;
    int lane = threadIdx.x & 31;
    if (wave >= B_ROWS) return;
    const float* r = logits + (size_t)wave * C_DIM;
    float x[C_DIM / 32];
    float m = -INFINITY;
#pragma unroll
    for (int t = 0; t < C_DIM / 32; ++t) { x[t] = r[lane + 32 * t]; m = fmaxf(m, x[t]); }
    for (int off = 16; off >= 1; off >>= 1) m = fmaxf(m, __shfl_xor(m, off, 32));
    float s = 0.f;
#pragma unroll
    for (int t = 0; t < C_DIM / 32; ++t) s += expf(x[t] - m);
    for (int off = 16; off >= 1; off >>= 1) s += __shfl_xor(s, off, 32);
    float lse = m + logf(s);
    if (lane == 0) atomicAdd(ceAcc, lse - r[labels[wave]]);
}

// ---------------------------------------------------------------------------
// Final scalar:
//   S = (1/D) * [ -2*sum_c HS_c*(2n_c-1-B) - 2*(2*sum_c P_c.L_c - Ptot.Ltot + sum H) ]
//   loss = 0.5*S/count + ce/B
// ---------------------------------------------------------------------------
__global__ __launch_bounds__(256) void finalize_kernel(
    const float* __restrict__ Pg, const float* __restrict__ Lg,
    const float* __restrict__ HSg, const float* __restrict__ CNTg,
    const float* __restrict__ ceAcc, float* __restrict__ lossOut) {
    __shared__ double red[256];
    double dotPL = 0.0, ptl = 0.0;
    for (int d = threadIdx.x; d < D_DIM; d += 256) {
        double pt = 0.0, lt = 0.0, pl = 0.0;
        for (int c = 0; c < NCLS; ++c) {
            double p = (double)Pg[c * D_DIM + d];
            double l = (double)Lg[c * D_DIM + d];
            pt += p; lt += l; pl += p * l;
        }
        dotPL += pl; ptl += pt * lt;
    }
    red[threadIdx.x] = dotPL; __syncthreads();
    for (int s = 128; s > 0; s >>= 1) { if (threadIdx.x < s) red[threadIdx.x] += red[threadIdx.x + s]; __syncthreads(); }
    double dotAll = red[0]; __syncthreads();
    red[threadIdx.x] = ptl; __syncthreads();
    for (int s = 128; s > 0; s >>= 1) { if (threadIdx.x < s) red[threadIdx.x] += red[threadIdx.x + s]; __syncthreads(); }
    if (threadIdx.x == 0) {
        double ptlAll = red[0];
        double sumH = 0.0, termA = 0.0;
        for (int c = 0; c < NCLS; ++c) {
            double h = (double)HSg[c], n = (double)CNTg[c];
            sumH  += h;
            termA += h * (2.0 * n - 1.0 - (double)B_ROWS);
        }
        double signM = 2.0 * dotAll - ptlAll + sumH;   // sum_{i!=j} sign_ij * M_ij
        double S = (-2.0 * termA - 2.0 * signM) / (double)D_DIM;
        double count = 0.5 * (double)B_ROWS * (double)(B_ROWS - 1);
        double loss_pair = 0.5 * S / count;
        double ce = (double)ceAcc[0] / (double)B_ROWS;
        lossOut[0] = (float)(loss_pair + ce);
    }
}

// ---------------------------------------------------------------------------
// Launcher
// ---------------------------------------------------------------------------
extern "C" void kernel_launch(void* const* d_in, const int* in_sizes, int n_in,
                              void* d_out, int out_size, void* d_ws, size_t ws_size,
                              hipStream_t stream) {
    const float* embeds = (const float*)d_in[0];
    const int*   labels = (const int*)d_in[1];
    const float* W1     = (const float*)d_in[2];
    const float* b1     = (const float*)d_in[3];
    const float* W2     = (const float*)d_in[4];
    const float* b2     = (const float*)d_in[5];

    float* out       = (float*)d_out;
    float* logitsOut = out;                                   // [4096,128]
    float* embedsOut = out + (size_t)B_ROWS * C_DIM;          // [4096,768]
    float* lossOut   = out + (size_t)B_ROWS * C_DIM + (size_t)B_ROWS * D_DIM;

    float* ws   = (float*)d_ws;
    float* h    = ws;                                         // [4096,768]
    float* Pg   = ws + (size_t)B_ROWS * D_DIM;                // [8,768]
    float* Lg   = Pg + NCLS * D_DIM;                          // [8,768]
    float* HSg  = Lg + NCLS * D_DIM;                          // [8]
    float* CNTg = HSg + NCLS;                                 // [8]
    float* ceAcc = CNTg + NCLS;                               // [1]

    // 1. zero the accumulators (P, L, HS, CNT, ce)
    int nz = 2 * NCLS * D_DIM + 2 * NCLS + 1;
    zero_kernel<<<(nz + 255) / 256, 256, 0, stream>>>(Pg, nz);

    // 2. embeds passthrough output
    hipMemcpyAsync(embedsOut, embeds, (size_t)B_ROWS * D_DIM * sizeof(float),
                   hipMemcpyDeviceToDevice, stream);

    // 3. h = relu(embeds @ W1^T + b1): blocks = (4096/128)*(768/64) = 384
    wmma_gemm_lds<true><<<(B_ROWS / 128) * (D_DIM / 64), 256, 0, stream>>>(
        embeds, W1, b1, h, B_ROWS, D_DIM, D_DIM);
    // 4. logits = h @ W2^T + b2: blocks = (4096/128)*(128/64) = 64
    wmma_gemm_lds<false><<<(B_ROWS / 128) * (C_DIM / 64), 256, 0, stream>>>(
        h, W2, b2, logitsOut, B_ROWS, C_DIM, D_DIM);
    // 5. softmax stats over embeds: 64 blocks x 64 rows
    stats_kernel<<<B_ROWS / 64, 256, 0, stream>>>(embeds, labels, Pg, Lg, HSg, CNTg);
    // 6. cross-entropy over logits: one wave per row
    ce_kernel<<<(B_ROWS * 32) / 256, 256, 0, stream>>>(logitsOut, labels, ceAcc);
    // 7. final scalar
    finalize_kernel<<<1, 256, 0, stream>>>(Pg, Lg, HSg, CNTg, ceAcc, lossOut);
}